// gat_62740882260600
// MI455X (gfx1250) — compile-verified
//
#include <hip/hip_runtime.h>
#include <math.h>

typedef __attribute__((ext_vector_type(2))) float v2f;
typedef __attribute__((ext_vector_type(8))) float v8f;

#define HEADS 3
#define OUTD 12
#define HID 36            // HEADS*OUTD
#define NEG_SLOPE 0.2f

// ---------------------------------------------------------------------------
// Dense node transform: h[N,HID] = yin[N,Fin] @ W[HID,Fin]^T
// via V_WMMA_F32_16X16X4_F32.  Block = 192 threads = 6 waves:
// 2 node tiles (16 rows each) x 3 output-column tiles (16 cols, 48 >= 36).
// ---------------------------------------------------------------------------
__launch_bounds__(192)
__global__ void gemm_wmma_kernel(const float* __restrict__ yin,
                                 const float* __restrict__ W,
                                 float* __restrict__ h,
                                 int N, int Fin) {
    __shared__ float yl[32][40];   // node tile, K padded
    __shared__ float wt[40][48];   // W^T tile: wt[k][out], cols padded to 48

    const int tid  = threadIdx.x;
    const int base = blockIdx.x * 32;

    for (int i = tid; i < 32 * 40; i += 192) ((float*)yl)[i] = 0.f;
    for (int i = tid; i < 40 * 48; i += 192) ((float*)wt)[i] = 0.f;
    __syncthreads();

    for (int i = tid; i < 32 * Fin; i += 192) {
        int r = i / Fin, k = i - r * Fin;
        int node = base + r;
        yl[r][k] = (node < N) ? yin[(size_t)node * Fin + k] : 0.f;
    }
    for (int i = tid; i < HID * Fin; i += 192) {
        int o = i / Fin, k = i - o * Fin;
        wt[k][o] = W[i];          // W is [HID, Fin] row-major -> transpose
    }
    __syncthreads();

    const int wave = tid >> 5;
    const int lane = tid & 31;
    const int hf   = lane >> 4;   // half-wave select
    const int lrow = lane & 15;
    const int mt   = wave / 3;           // node tile 0..1
    const int nt   = wave - mt * 3;      // col tile 0..2
    const int colb = nt * 16;

    v8f c = {};
    for (int kk = 0; kk < Fin; kk += 4) {
        // A 16x4 f32 layout: lanes 0-15 hold K=0,1 ; lanes 16-31 hold K=2,3
        v2f a, b;
        a.x = yl[mt * 16 + lrow][kk + 2 * hf + 0];
        a.y = yl[mt * 16 + lrow][kk + 2 * hf + 1];
        // B 4x16 f32 layout: VGPR0 rows K=0/K=2, VGPR1 rows K=1/K=3
        b.x = wt[kk + 2 * hf + 0][colb + lrow];
        b.y = wt[kk + 2 * hf + 1][colb + lrow];
        c = __builtin_amdgcn_wmma_f32_16x16x4_f32(false, a, false, b,
                                                  (short)0, c, false, false);
    }

    // D 16x16 f32: VGPR j -> M=j (lanes 0-15), M=j+8 (lanes 16-31), N=lane%16
    #pragma unroll
    for (int j = 0; j < 8; ++j) {
        int row = base + mt * 16 + j + 8 * hf;
        int col = colb + lrow;
        if (row < N && col < HID) h[(size_t)row * HID + col] = c[j];
    }
}

// ---------------------------------------------------------------------------
// Per-node: attention logits al_src/al_dst, init segment max/sum, zero agg.
// ---------------------------------------------------------------------------
__global__ void node_prep_kernel(const float* __restrict__ h,
                                 const float* __restrict__ a_src,
                                 const float* __restrict__ a_dst,
                                 float* __restrict__ als, float* __restrict__ ald,
                                 float* __restrict__ mbuf, float* __restrict__ sbuf,
                                 float* __restrict__ agg, int N) {
    int n = blockIdx.x * blockDim.x + threadIdx.x;
    if (n >= N) return;
    const float* hp = h + (size_t)n * HID;
    #pragma unroll
    for (int hd = 0; hd < HEADS; ++hd) {
        float ss = 0.f, dd = 0.f;
        #pragma unroll
        for (int o = 0; o < OUTD; ++o) {
            float v = hp[hd * OUTD + o];
            ss += v * a_src[hd * OUTD + o];
            dd += v * a_dst[hd * OUTD + o];
        }
        als[n * HEADS + hd]  = ss;
        ald[n * HEADS + hd]  = dd;
        mbuf[n * HEADS + hd] = -INFINITY;
        sbuf[n * HEADS + hd] = 0.f;
    }
    float* ap = agg + (size_t)n * HID;
    #pragma unroll
    for (int k = 0; k < HID; ++k) ap[k] = 0.f;
}

// float atomic max via ordered-int trick (no CAS loop; lowers to
// global_atomic_max_i32 / global_atomic_min_u32 in L2)
__device__ __forceinline__ void atomicMaxF(float* addr, float val) {
    if (val >= 0.f) atomicMax((int*)addr, __float_as_int(val));
    else            atomicMin((unsigned int*)addr, (unsigned int)__float_as_int(val));
}

__device__ __forceinline__ void edge_pair(int e, int E,
                                          const int* __restrict__ src,
                                          const int* __restrict__ dst,
                                          int& s, int& d) {
    if (e < E) { s = src[e]; d = dst[e]; }
    else       { s = e - E;  d = e - E; }     // appended self-loops
}

__global__ void edge_max_kernel(const int* __restrict__ src, const int* __restrict__ dst,
                                const float* __restrict__ als, const float* __restrict__ ald,
                                float* __restrict__ mbuf, int E, int Etot) {
    int e = blockIdx.x * blockDim.x + threadIdx.x;
    if (e >= Etot) return;
    int s, d; edge_pair(e, E, src, dst, s, d);
    #pragma unroll
    for (int hd = 0; hd < HEADS; ++hd) {
        float v = als[s * HEADS + hd] + ald[d * HEADS + hd];
        v = (v > 0.f) ? v : NEG_SLOPE * v;
        atomicMaxF(&mbuf[d * HEADS + hd], v);
    }
}

__global__ void edge_sum_kernel(const int* __restrict__ src, const int* __restrict__ dst,
                                const float* __restrict__ als, const float* __restrict__ ald,
                                const float* __restrict__ mbuf, float* __restrict__ sbuf,
                                int E, int Etot) {
    int e = blockIdx.x * blockDim.x + threadIdx.x;
    if (e >= Etot) return;
    int s, d; edge_pair(e, E, src, dst, s, d);
    #pragma unroll
    for (int hd = 0; hd < HEADS; ++hd) {
        float v = als[s * HEADS + hd] + ald[d * HEADS + hd];
        v = (v > 0.f) ? v : NEG_SLOPE * v;
        float mm = mbuf[d * HEADS + hd];
        if (mm == -INFINITY) mm = 0.f;          // empty-segment guard
        atomicAdd(&sbuf[d * HEADS + hd], __expf(v - mm));
    }
}

__global__ void edge_agg_kernel(const int* __restrict__ src, const int* __restrict__ dst,
                                const float* __restrict__ als, const float* __restrict__ ald,
                                const float* __restrict__ mbuf, const float* __restrict__ sbuf,
                                const float* __restrict__ h, float* __restrict__ agg,
                                int E, int Etot) {
    int e = blockIdx.x * blockDim.x + threadIdx.x;
    if (e >= Etot) return;
    int s, d; edge_pair(e, E, src, dst, s, d);
    #pragma unroll
    for (int hd = 0; hd < HEADS; ++hd) {
        float v = als[s * HEADS + hd] + ald[d * HEADS + hd];
        v = (v > 0.f) ? v : NEG_SLOPE * v;
        float mm = mbuf[d * HEADS + hd];
        if (mm == -INFINITY) mm = 0.f;
        float alpha = __expf(v - mm) / (sbuf[d * HEADS + hd] + 1e-16f);
        const float4* hp = (const float4*)(h + (size_t)s * HID + hd * OUTD);
        float* ap = agg + (size_t)d * HID + hd * OUTD;
        #pragma unroll
        for (int q = 0; q < 3; ++q) {
            float4 hv = hp[q];                 // 128-bit gather from L2-resident h
            atomicAdd(ap + q * 4 + 0, hv.x * alpha);
            atomicAdd(ap + q * 4 + 1, hv.y * alpha);
            atomicAdd(ap + q * 4 + 2, hv.z * alpha);
            atomicAdd(ap + q * 4 + 3, hv.w * alpha);
        }
    }
}

__global__ void finalize_relu_kernel(const float* __restrict__ agg,
                                     const float* __restrict__ bias,
                                     float* __restrict__ yout, int total) {
    int i = blockIdx.x * blockDim.x + threadIdx.x;
    if (i >= total) return;
    float v = agg[i] + bias[i % HID];
    yout[i] = (v > 0.f) ? v : 0.f;
}

// layer 3: mean over heads + b3, then lin1 (12x12) and lin2 (6x12)
__global__ void final_head_kernel(const float* __restrict__ agg,
                                  const float* __restrict__ b3,
                                  const float* __restrict__ lin1_w,
                                  const float* __restrict__ lin1_b,
                                  const float* __restrict__ lin2_w,
                                  const float* __restrict__ lin2_b,
                                  float* __restrict__ out, int N) {
    int n = blockIdx.x * blockDim.x + threadIdx.x;
    if (n >= N) return;
    const float* ap = agg + (size_t)n * HID;
    float v[OUTD], t[OUTD];
    #pragma unroll
    for (int o = 0; o < OUTD; ++o)
        v[o] = (ap[o] + ap[OUTD + o] + ap[2 * OUTD + o]) * (1.f / 3.f) + b3[o];
    #pragma unroll
    for (int i = 0; i < 12; ++i) {
        float acc = lin1_b[i];
        #pragma unroll
        for (int j = 0; j < 12; ++j) acc += lin1_w[i * 12 + j] * v[j];
        t[i] = acc;
    }
    #pragma unroll
    for (int i = 0; i < 6; ++i) {
        float acc = lin2_b[i];
        #pragma unroll
        for (int j = 0; j < 12; ++j) acc += lin2_w[i * 12 + j] * t[j];
        out[(size_t)n * 6 + i] = acc;
    }
}

// ---------------------------------------------------------------------------
extern "C" void kernel_launch(void* const* d_in, const int* in_sizes, int n_in,
                              void* d_out, int out_size, void* d_ws, size_t ws_size,
                              hipStream_t stream) {
    const float* x  = (const float*)d_in[0];      // [1,N,12,2] == [N,24]
    const int*   ei = (const int*)d_in[1];        // [2,E]
    const int E    = in_sizes[1] / 2;
    const int N    = in_sizes[0] / 24;
    const int Etot = E + N;                       // + self loops
    const int* srcp = ei;
    const int* dstp = ei + E;

    float* w      = (float*)d_ws;
    float* bufA   = w;                            // [N,36] layer input (l>=1)
    float* bufH   = bufA   + (size_t)N * HID;     // [N,36] transformed h
    float* bufAgg = bufH   + (size_t)N * HID;     // [N,36] aggregation
    float* als    = bufAgg + (size_t)N * HID;     // [N,3]
    float* ald    = als    + (size_t)N * HEADS;
    float* mbuf   = ald    + (size_t)N * HEADS;
    float* sbuf   = mbuf   + (size_t)N * HEADS;

    const int gN  = (N + 255) / 256;
    const int gE  = (Etot + 255) / 256;
    const int gG  = (N + 31) / 32;
    const int gNH = (N * HID + 255) / 256;

    for (int l = 0; l < 4; ++l) {
        const float* W  = (const float*)d_in[2 + 4 * l];
        const float* as = (const float*)d_in[3 + 4 * l];
        const float* ad = (const float*)d_in[4 + 4 * l];
        const float* bb = (const float*)d_in[5 + 4 * l];
        const int Fin = (l == 0) ? 24 : HID;
        const float* yin = (l == 0) ? x : bufA;

        gemm_wmma_kernel<<<gG, 192, 0, stream>>>(yin, W, bufH, N, Fin);
        node_prep_kernel<<<gN, 256, 0, stream>>>(bufH, as, ad, als, ald,
                                                 mbuf, sbuf, bufAgg, N);
        edge_max_kernel<<<gE, 256, 0, stream>>>(srcp, dstp, als, ald, mbuf, E, Etot);
        edge_sum_kernel<<<gE, 256, 0, stream>>>(srcp, dstp, als, ald, mbuf, sbuf, E, Etot);
        edge_agg_kernel<<<gE, 256, 0, stream>>>(srcp, dstp, als, ald, mbuf, sbuf,
                                                bufH, bufAgg, E, Etot);
        if (l < 3) {
            finalize_relu_kernel<<<gNH, 256, 0, stream>>>(bufAgg, bb, bufA, N * HID);
        } else {
            final_head_kernel<<<gN, 256, 0, stream>>>(bufAgg, bb,
                (const float*)d_in[18], (const float*)d_in[19],
                (const float*)d_in[20], (const float*)d_in[21],
                (float*)d_out, N);
        }
    }
}